// OptModel_YW_22917945491919
// MI455X (gfx1250) — compile-verified
//
#include <hip/hip_runtime.h>

// ---------------- problem constants ----------------
#define NJ      25        // joints
#define NDOF    78        // 3 + 25*3
#define NL      15        // tracked links (L_B)
#define NRES    135       // env(45) + up(45) + fwd(45); res-block folded analytically
#define KPAD    136       // NRES padded to multiple of 4 for WMMA K loop
#define JS      80        // J column stride (78 dof padded to 80 = 5 WMMA tiles)
#define JTS     81        // JTJ row stride (81: 81*i mod 64 is a permutation -> no LDS bank conflicts)
#define NITER   15
#define STEPSZ  0.7f
#define DAMP    0.001f

__constant__ int c_PAR[NJ] = {-1,0,1,2,3,1,5,6,7,1,9,10,11,0,13,14,15,0,17,18,19,3,21,3,23};
__constant__ int c_LB[NL]  = {0,1,3,4,6,8,9,13,15,16,17,20,22,23,24};
// lower-triangular 5x5 tile enumeration for symmetric JTJ
__constant__ int c_TI[15]  = {0,1,1,2,2,2,3,3,3,3,4,4,4,4,4};
__constant__ int c_TJ[15]  = {0,0,1,0,1,2,0,1,2,3,0,1,2,3,4};

typedef float v2f __attribute__((ext_vector_type(2)));
typedef float v8f __attribute__((ext_vector_type(8)));

// R = I + sinc(|w|) K + cosc(|w|) (w w^T - |w|^2 I),  K = skew(w)
__device__ __forceinline__ void exp_so3_primal(const float w[3], float R[3][3]) {
    float a2 = w[0]*w[0] + w[1]*w[1] + w[2]*w[2];
    float sinc, cosc;
    if (a2 < 1e-8f) {
        sinc = 1.f - a2 * (1.f/6.f);
        cosc = 0.5f - a2 * (1.f/24.f);
    } else {
        float a = sqrtf(a2);
        float s = sinf(a), c = cosf(a);
        sinc = s / a;
        cosc = (1.f - c) / a2;
    }
    float K[3][3] = {{0.f,-w[2],w[1]},{w[2],0.f,-w[0]},{-w[1],w[0],0.f}};
#pragma unroll
    for (int i = 0; i < 3; ++i)
#pragma unroll
        for (int j = 0; j < 3; ++j) {
            float kk = w[i]*w[j] - (i==j ? a2 : 0.f);
            R[i][j] = (i==j ? 1.f : 0.f) + sinc*K[i][j] + cosc*kk;
        }
}

// dR = d exp_so3(w) / d w[cd]   (directional derivative along basis vector cd)
__device__ __forceinline__ void exp_so3_dexp(const float w[3], int cd, float dR[3][3]) {
    float a2  = w[0]*w[0] + w[1]*w[1] + w[2]*w[2];
    float da2 = 2.f * w[cd];
    float sinc, cosc, dsinc, dcosc;
    if (a2 < 1e-8f) {
        sinc  = 1.f - a2 * (1.f/6.f);   dsinc = -da2 * (1.f/6.f);
        cosc  = 0.5f - a2 * (1.f/24.f); dcosc = -da2 * (1.f/24.f);
    } else {
        float a = sqrtf(a2);
        float s = sinf(a), c = cosf(a);
        float inva = 1.f / a, inva2 = 1.f / a2;
        sinc = s * inva;
        cosc = (1.f - c) * inva2;
        float da = da2 * 0.5f * inva;
        dsinc = (c*inva - s*inva2) * da;                       // d(sin a / a)/da
        dcosc = ((a*s - 2.f*(1.f - c)) * inva2 * inva) * da;   // d((1-cos a)/a^2)/da
    }
    float e[3] = {(cd==0)?1.f:0.f, (cd==1)?1.f:0.f, (cd==2)?1.f:0.f};
    float K [3][3] = {{0.f,-w[2],w[1]},{w[2],0.f,-w[0]},{-w[1],w[0],0.f}};
    float dK[3][3] = {{0.f,-e[2],e[1]},{e[2],0.f,-e[0]},{-e[1],e[0],0.f}};
#pragma unroll
    for (int i = 0; i < 3; ++i)
#pragma unroll
        for (int j = 0; j < 3; ++j) {
            float kk  = w[i]*w[j] - (i==j ? a2  : 0.f);
            float dkk = e[i]*w[j] + w[i]*e[j] - (i==j ? da2 : 0.f);
            dR[i][j] = dsinc*K[i][j] + sinc*dK[i][j] + dcosc*kk + cosc*dkk;
        }
}

// HIP launch_bounds: (max threads/block, min waves per EU) -> cap at 256 VGPRs/wave
__global__ __launch_bounds__(256, 4) void ik_lm_kernel(
    const float* __restrict__ theta0, const float* __restrict__ kp,
    const float* __restrict__ ku,     const float* __restrict__ kf,
    const float* __restrict__ q,      const float* __restrict__ off,
    float* __restrict__ out)
{
    __shared__ float sJ[KPAD * JS];       // residual-major Jacobian, zero-padded
    __shared__ float sJTJ[JS * JTS];
    __shared__ float sr[KPAD];
    __shared__ float srhs[JS], sx[JS];
    __shared__ float sth[JS], sq[JS];
    __shared__ float skp[48], sku[48], skf[48];
    __shared__ float soff[NJ*3 + 1];
    __shared__ float sRj[NJ*9];           // per-joint local rotations
    __shared__ float sRw[NJ*9];           // world rotations
    __shared__ float spw[NJ*3 + 1];       // world positions
    __shared__ int   sAnc[NL + 1];        // ancestor-or-self bitmask per tracked link

    const int blk = blockIdx.x;
    const int tid = threadIdx.x;

    // ---- one-time loads ----
    if (tid < NDOF) {
        sth[tid] = theta0[(size_t)blk * NDOF + tid];
        sq[tid]  = q     [(size_t)blk * NDOF + tid];
    }
    if (tid < 45) {
        skp[tid] = kp[(size_t)blk * 45 + tid];
        sku[tid] = ku[(size_t)blk * 45 + tid];
        skf[tid] = kf[(size_t)blk * 45 + tid];
    }
    if (tid < NJ*3) soff[tid] = off[tid];
    if (tid < NL) {
        int m = 0, i = c_LB[tid];
        while (i >= 0) { m |= (1 << i); i = c_PAR[i]; }
        sAnc[tid] = m;
    }
    // Zero J once: structurally-zero entries (pad cols 78-79, pad row 135,
    // non-ancestor link rows, translation off-diagonals) are never rewritten.
    for (int i = tid; i < KPAD*JS; i += 256) sJ[i] = 0.f;
    __syncthreads();

    for (int it = 0; it < NITER; ++it) {
        // ---- 1) per-joint local rotations (25 threads) ----
        if (tid < NJ) {
            float w[3] = { sth[3 + tid*3 + 0], sth[3 + tid*3 + 1], sth[3 + tid*3 + 2] };
            float R[3][3];
            exp_so3_primal(w, R);
#pragma unroll
            for (int r = 0; r < 3; ++r)
#pragma unroll
                for (int c = 0; c < 3; ++c) sRj[tid*9 + r*3 + c] = R[r][c];
        }
        __syncthreads();

        // ---- 2) serial FK chain (tiny: ~1.3 kFLOP) ----
        if (tid == 0) {
#pragma unroll
            for (int k = 0; k < 9; ++k) sRw[k] = sRj[k];
#pragma unroll
            for (int k = 0; k < 3; ++k) spw[k] = sth[k] + soff[k];
#pragma unroll 1
            for (int i = 1; i < NJ; ++i) {
                int p = c_PAR[i];
#pragma unroll
                for (int r = 0; r < 3; ++r) {
#pragma unroll
                    for (int c = 0; c < 3; ++c) {
                        float acc = 0.f;
#pragma unroll
                        for (int k = 0; k < 3; ++k) acc += sRw[p*9 + r*3 + k] * sRj[i*9 + k*3 + c];
                        sRw[i*9 + r*3 + c] = acc;
                    }
                    float pv = spw[p*3 + r];
#pragma unroll
                    for (int k = 0; k < 3; ++k) pv += sRw[p*9 + r*3 + k] * soff[i*3 + k];
                    spw[i*3 + r] = pv;
                }
            }
        }
        __syncthreads();

        // ---- 3) residual + analytic Jacobian columns ----
        if (tid < NRES) {
            int k = tid; float v;
            if (k < 45)      { int s=k/3,      c=k%3,      l=c_LB[s]; v = fabsf(spw[l*3+c]       - skp[k]);  }
            else if (k < 90) { int kk=k-45, s=kk/3, c=kk%3, l=c_LB[s]; v = fabsf(sRw[l*9+c*3+1] - sku[kk]); }
            else             { int kk=k-90, s=kk/3, c=kk%3, l=c_LB[s]; v = fabsf(sRw[l*9+c*3+2] - skf[kk]); }
            sr[k] = v;
        }
        if (tid < NDOF) {
            const int d = tid;
            if (d < 3) {   // translation columns: only diag entries of each env triple are nonzero
#pragma unroll 1
                for (int s = 0; s < NL; ++s) {
                    int l = c_LB[s];
                    sJ[(s*3+d)*JS + d] = copysignf(1.f, spw[l*3+d] - skp[s*3+d]);
                }
            } else {
                const int jd = (d-3)/3, cd = (d-3)%3;
                float w[3] = { sth[3 + jd*3 + 0], sth[3 + jd*3 + 1], sth[3 + jd*3 + 2] };
                float dR[3][3];
                exp_so3_dexp(w, cd, dR);
                // A = Rw[parent(jd)] * dR   (identity parent for the root)
                float A[3][3];
                int p = c_PAR[jd];
                if (p < 0) {
#pragma unroll
                    for (int r = 0; r < 3; ++r)
#pragma unroll
                        for (int c = 0; c < 3; ++c) A[r][c] = dR[r][c];
                } else {
#pragma unroll
                    for (int r = 0; r < 3; ++r)
#pragma unroll
                        for (int c = 0; c < 3; ++c) {
                            float acc = 0.f;
#pragma unroll
                            for (int k = 0; k < 3; ++k) acc += sRw[p*9 + r*3 + k] * dR[k][c];
                            A[r][c] = acc;
                        }
                }
                // B3 = A * Rw[jd]^T  => dRw[l] = B3*Rw[l], dpw[l] = B3*(pw[l]-pw[jd]) for descendants l
                float B3[3][3];
#pragma unroll
                for (int r = 0; r < 3; ++r)
#pragma unroll
                    for (int c = 0; c < 3; ++c) {
                        float acc = 0.f;
#pragma unroll
                        for (int k = 0; k < 3; ++k) acc += A[r][k] * sRw[jd*9 + c*3 + k];
                        B3[r][c] = acc;
                    }
#pragma unroll 1
                for (int s = 0; s < NL; ++s) {
                    if (!((sAnc[s] >> jd) & 1)) continue;   // structurally zero rows stay 0 in LDS
                    int l = c_LB[s];
                    float r0 = spw[l*3+0]-spw[jd*3+0], r1 = spw[l*3+1]-spw[jd*3+1], r2 = spw[l*3+2]-spw[jd*3+2];
                    float u0 = sRw[l*9+1], u1 = sRw[l*9+4], u2 = sRw[l*9+7];   // Rw col 1
                    float f0 = sRw[l*9+2], f1 = sRw[l*9+5], f2 = sRw[l*9+8];   // Rw col 2
#pragma unroll
                    for (int c = 0; c < 3; ++c) {
                        float dp = B3[c][0]*r0 + B3[c][1]*r1 + B3[c][2]*r2;
                        float du = B3[c][0]*u0 + B3[c][1]*u1 + B3[c][2]*u2;
                        float df = B3[c][0]*f0 + B3[c][1]*f1 + B3[c][2]*f2;
                        sJ[(s*3+c)*JS + d]    = copysignf(1.f, spw[l*3+c]     - skp[s*3+c]) * dp;
                        sJ[(45+s*3+c)*JS + d] = copysignf(1.f, sRw[l*9+c*3+1] - sku[s*3+c]) * du;
                        sJ[(90+s*3+c)*JS + d] = copysignf(1.f, sRw[l*9+c*3+2] - skf[s*3+c]) * df;
                    }
                }
            }
        }
        __syncthreads();

        // ---- 4a) g = J^T r (+ analytic c_res term), rhs = -g ----
        if (tid < NDOF) {
            float acc = 0.f;
#pragma unroll 1
            for (int k = 0; k < NRES; ++k) acc += sJ[k*JS + tid] * sr[k];
            if (tid >= 3) acc += 0.01f * (sth[tid] - sq[tid]);   // folded c_res gradient
            srhs[tid] = -acc;
        }
        // ---- 4b) JTJ = J^T J via V_WMMA_F32_16X16X4_F32, symmetric lower tiles only ----
        {
            const int lane = tid & 31, wv = tid >> 5;
            const int m = lane & 15, half = lane >> 4;
            for (int t = wv; t < 15; t += 8) {            // wave-uniform trip count: EXEC all-1s
                const int ts = __builtin_amdgcn_readfirstlane(t);   // force scalar tile indices
                const int ti = c_TI[ts], tj = c_TJ[ts];
                v8f acc = {0.f,0.f,0.f,0.f,0.f,0.f,0.f,0.f};
#pragma unroll 2
                for (int k = 0; k < KPAD; k += 4) {
                    const int kh = k + 2*half;            // A/B VGPR0->K=2*half, VGPR1->K=2*half+1
                    v2f a, b;
                    a.x = sJ[(kh  )*JS + ti*16 + m];      // A = (J^T) tile: A[m][kk] = J[k+kk][ti*16+m]
                    a.y = sJ[(kh+1)*JS + ti*16 + m];
                    b.x = sJ[(kh  )*JS + tj*16 + m];      // B = J tile:     B[kk][n] = J[k+kk][tj*16+n]
                    b.y = sJ[(kh+1)*JS + tj*16 + m];
                    acc = __builtin_amdgcn_wmma_f32_16x16x4_f32(
                        false, a, false, b, (short)0, acc, false, false);
                }
                const int row0 = ti*16 + 8*half, col = tj*16 + m;  // C layout: VGPR r -> M=r(+8)
                if (ti != tj) {                                    // scalar branch, no EXEC churn
#pragma unroll
                    for (int rr = 0; rr < 8; ++rr) {
                        sJTJ[(row0+rr)*JTS + col] = acc[rr];
                        sJTJ[col*JTS + (row0+rr)] = acc[rr];       // symmetric mirror
                    }
                } else {
#pragma unroll
                    for (int rr = 0; rr < 8; ++rr)
                        sJTJ[(row0+rr)*JTS + col] = acc[rr];
                }
            }
        }
        __syncthreads();

        // ---- 5) damping + folded c_res diagonal ----
        if (tid < NDOF) sJTJ[tid*JTS + tid] += DAMP + (tid >= 3 ? 0.01f : 0.f);
        __syncthreads();

        // ---- 6) Gaussian elimination (thread-per-row) ----
#pragma unroll 1
        for (int k = 0; k < NDOF-1; ++k) {
            if (tid > k && tid < NDOF) {
                float f = sJTJ[tid*JTS + k] / sJTJ[k*JTS + k];
#pragma unroll 1
                for (int j = k+1; j < NDOF; ++j) sJTJ[tid*JTS + j] -= f * sJTJ[k*JTS + j];
                srhs[tid] -= f * srhs[k];
            }
            __syncthreads();
        }
        // ---- 7) back substitution ----
#pragma unroll 1
        for (int k = NDOF-1; k >= 0; --k) {
            if (tid == k) sx[k] = srhs[k] / sJTJ[k*JTS + k];
            __syncthreads();
            if (tid < k) srhs[tid] -= sJTJ[tid*JTS + k] * sx[k];
            __syncthreads();
        }
        // ---- 8) step ----
        if (tid < NDOF) sth[tid] += STEPSZ * sx[tid];
        __syncthreads();
    }

    if (tid < NDOF) out[(size_t)blk * NDOF + tid] = sth[tid];
}

extern "C" void kernel_launch(void* const* d_in, const int* in_sizes, int n_in,
                              void* d_out, int out_size, void* d_ws, size_t ws_size,
                              hipStream_t stream) {
    const float* theta0 = (const float*)d_in[0];
    const float* kp     = (const float*)d_in[1];
    const float* ku     = (const float*)d_in[2];
    const float* kf     = (const float*)d_in[3];
    const float* q      = (const float*)d_in[4];
    const float* off    = (const float*)d_in[5];
    const int N = in_sizes[0] / NDOF;   // 4096 independent problems, one workgroup each
    ik_lm_kernel<<<N, 256, 0, stream>>>(theta0, kp, ku, kf, q, off, (float*)d_out);
}